// GNNConv_12850542149846
// MI455X (gfx1250) — compile-verified
//
#include <hip/hip_runtime.h>

#define N_NODES 100000
#define N_EDGES 1000000
#define D 64

typedef float v2f __attribute__((ext_vector_type(2)));
typedef float v8f __attribute__((ext_vector_type(8)));

// ---------------------------------------------------------------------------
// Kernel 1: zero agg[N*64] and deg[N] (laid out contiguously in workspace)
// ---------------------------------------------------------------------------
__global__ void sage_zero_kernel(float* __restrict__ p, int n) {
    int t = blockIdx.x * blockDim.x + threadIdx.x;
    if (t < n) p[t] = 0.0f;
}

// ---------------------------------------------------------------------------
// Kernel 2: edge scatter. 16 threads per edge; each thread moves one float4
// of x[src] into agg[dst] via f32 global atomics (resolved in L2).
// Group 0 also accumulates the degree.
// ---------------------------------------------------------------------------
__global__ void sage_scatter_kernel(const float* __restrict__ x,
                                    const int* __restrict__ src,
                                    const int* __restrict__ dst,
                                    float* __restrict__ agg,
                                    float* __restrict__ deg) {
    int t = blockIdx.x * blockDim.x + threadIdx.x;   // [0, N_EDGES*16)
    if (t >= N_EDGES * 16) return;
    int e = t >> 4;
    int g = t & 15;
    int s = src[e];
    int i = dst[e];
    const float4 v = ((const float4*)x)[s * (D / 4) + g];
    float* a = agg + i * D + g * 4;
    atomicAdd(a + 0, v.x);
    atomicAdd(a + 1, v.y);
    atomicAdd(a + 2, v.z);
    atomicAdd(a + 3, v.w);
    if (g == 0) atomicAdd(&deg[i], 1.0f);
}

// ---------------------------------------------------------------------------
// Kernel 3: fused  out = relu( (agg/max(deg,1)) @ W_l + b_l + x @ W_r )
// One wave32 computes one 16x16 output tile with V_WMMA_F32_16X16X4_F32.
//
// f32 16x4 A layout : lanes 0-15 hold row M=lane, VGPR0=K+0, VGPR1=K+1;
//                     lanes 16-31 hold row M=lane-16, VGPR0=K+2, VGPR1=K+3.
// f32 4x16 B layout : mirrored with lane = N.
// f32 16x16 C/D     : VGPR i -> M=i (lanes 0-15) / M=i+8 (lanes 16-31), N=lane&15.
//
// The mean-normalization uses ONE IEEE divide per lane (loop-invariant) and
// scales the 32 A elements with v_mul_f32, keeping the VALU stream light so
// it can co-issue with the WMMA (XDL) pipe instead of serializing behind
// 32 full v_div_* expansions.
// ---------------------------------------------------------------------------
__global__ void sage_wmma_gemm_kernel(const float* __restrict__ agg,
                                      const float* __restrict__ deg,
                                      const float* __restrict__ x,
                                      const float* __restrict__ Wl,
                                      const float* __restrict__ bl,
                                      const float* __restrict__ Wr,
                                      float* __restrict__ out) {
    const int lane  = threadIdx.x & 31;
    const int wave  = threadIdx.x >> 5;
    const int tile  = blockIdx.x * (blockDim.x >> 5) + wave;  // 25000 tiles total
    const int rowTile = tile >> 2;    // N_NODES/16 row tiles
    const int colTile = tile & 3;     // 64/16 col tiles
    const int row0 = rowTile * 16;
    const int col0 = colTile * 16;
    const int mrow  = lane & 15;      // M (for A) or N (for B/C/D) within tile
    const int khalf = lane >> 4;      // selects K+0/1 vs K+2/3 within a k-step

    // Per-lane mean scale: one IEEE divide, then multiplies (loop-invariant).
    const float rscale = 1.0f / fmaxf(deg[row0 + mrow], 1.0f);

    // Accumulator initialized with the bias (same value across all 8 M rows).
    const float bias = bl[col0 + mrow];
    v8f acc;
#pragma unroll
    for (int i = 0; i < 8; ++i) acc[i] = bias;

    const float* aBase = agg + (size_t)(row0 + mrow) * D;
    const float* xBase = x   + (size_t)(row0 + mrow) * D;

    // GEMM 1: (agg * rscale) @ W_l
#pragma unroll
    for (int k = 0; k < D; k += 4) {
        const int ka = k + khalf * 2;
        v2f a, b;
        a.x = aBase[ka]     * rscale;
        a.y = aBase[ka + 1] * rscale;
        b.x = Wl[ka       * D + col0 + mrow];
        b.y = Wl[(ka + 1) * D + col0 + mrow];
        acc = __builtin_amdgcn_wmma_f32_16x16x4_f32(false, a, false, b,
                                                    (short)0, acc, false, false);
    }
    // GEMM 2: x @ W_r
#pragma unroll
    for (int k = 0; k < D; k += 4) {
        const int ka = k + khalf * 2;
        v2f a, b;
        a.x = xBase[ka];
        a.y = xBase[ka + 1];
        b.x = Wr[ka       * D + col0 + mrow];
        b.y = Wr[(ka + 1) * D + col0 + mrow];
        acc = __builtin_amdgcn_wmma_f32_16x16x4_f32(false, a, false, b,
                                                    (short)0, acc, false, false);
    }

    // ReLU + store: VGPR i is row row0 + khalf*8 + i, column col0 + mrow.
    const int rbase = row0 + khalf * 8;
#pragma unroll
    for (int i = 0; i < 8; ++i) {
        out[(size_t)(rbase + i) * D + col0 + mrow] = fmaxf(acc[i], 0.0f);
    }
}

// ---------------------------------------------------------------------------
extern "C" void kernel_launch(void* const* d_in, const int* in_sizes, int n_in,
                              void* d_out, int out_size, void* d_ws, size_t ws_size,
                              hipStream_t stream) {
    const float* x  = (const float*)d_in[0];   // [N_NODES, D]
    const int*   ei = (const int*)d_in[1];     // [2, N_EDGES]
    const float* Wl = (const float*)d_in[2];   // [D, D]
    const float* bl = (const float*)d_in[3];   // [D]
    const float* Wr = (const float*)d_in[4];   // [D, D]
    float* out = (float*)d_out;                // [N_NODES, D]

    float* agg = (float*)d_ws;                     // N_NODES*D floats
    float* deg = agg + (size_t)N_NODES * D;        // N_NODES floats (contiguous)

    const int* src = ei;            // edge_index[0]
    const int* dst = ei + N_EDGES;  // edge_index[1]

    // 1) zero scratch (agg + deg contiguous)
    const int nz = N_NODES * D + N_NODES;
    sage_zero_kernel<<<(nz + 255) / 256, 256, 0, stream>>>(agg, nz);

    // 2) edge scatter (16 threads per edge)
    const int nscat = N_EDGES * 16;
    sage_scatter_kernel<<<(nscat + 255) / 256, 256, 0, stream>>>(x, src, dst, agg, deg);

    // 3) fused WMMA GEMM + bias + ReLU; 25000 tiles, 8 waves (tiles) per block
    const int nTiles = (N_NODES / 16) * (D / 16);  // 25000, divisible by 8
    sage_wmma_gemm_kernel<<<nTiles / 8, 256, 0, stream>>>(agg, deg, x, Wl, bl, Wr, out);
}